// PSRoIPool_58033598103664
// MI455X (gfx1250) — compile-verified
//
#include <hip/hip_runtime.h>
#include <hip/hip_bf16.h>

typedef __attribute__((ext_vector_type(2))) float v2f;
typedef __attribute__((ext_vector_type(8))) float v8f;

#define C_OUT   21
#define PHH     7
#define PWW     7
#define HH      96
#define WW      96
#define CIN     (C_OUT * PHH * PWW)   /* 1029 */
#define SCALE   0.0625f

// One wave32 per (roi, bin_i, bin_j). Channels ride the WMMA M dimension,
// bin pixels ride the K dimension (4 per v_wmma_f32_16x16x4_f32 step),
// B = all-ones so every D column accumulates per-channel bin sums in f32.
//
// Branch-free inner loop: loads are unconditional from clamped addresses;
// out-of-range K slots are zeroed with selects. Lanes whose group-1 channel
// would exceed 20 are clamped to plane 20 and accumulate into D rows 5..15,
// which are never extracted (WMMA rows are independent), so no mask needed.
__global__ __launch_bounds__(32)
void psroi_pool_wmma(const float* __restrict__ x,
                     const float* __restrict__ rois,
                     float* __restrict__ out)
{
    const int bin  = blockIdx.x;          // 0..48
    const int k    = blockIdx.y;          // roi index
    const int i    = bin / PWW;
    const int j    = bin % PWW;
    const int lane = threadIdx.x;

    // ROI -> feature-map rectangle (reference-exact rounding).
    const float r0  = rois[k * 5 + 0];
    const float rx1 = rois[k * 5 + 1];
    const float ry1 = rois[k * 5 + 2];
    const float rx2 = rois[k * 5 + 3];
    const float ry2 = rois[k * 5 + 4];

    const int b  = (int)r0;
    const int sw = (int)floorf(rx1 * SCALE + 0.5f);
    const int sh = (int)floorf(ry1 * SCALE + 0.5f);
    const int ew = (int)floorf(rx2 * SCALE + 0.5f);
    const int eh = (int)floorf(ry2 * SCALE + 0.5f);

    const float roi_w = fmaxf((float)(ew - sw), 1.0f);
    const float roi_h = fmaxf((float)(eh - sh), 1.0f);
    const float bin_h = roi_h * (1.0f / PHH);
    const float bin_w = roi_w * (1.0f / PWW);

    int hs = (int)floorf((float)i       * bin_h) + sh;
    int he = (int)ceilf ((float)(i + 1) * bin_h) + sh;
    int ws = (int)floorf((float)j       * bin_w) + sw;
    int we = (int)ceilf ((float)(j + 1) * bin_w) + sw;
    hs = min(max(hs, 0), HH);  he = min(max(he, 0), HH);
    ws = min(max(ws, 0), WW);  we = min(max(we, 0), WW);

    // Wave-uniform in value; force SGPR so loops are scalar branches and
    // EXEC is all-ones at every WMMA (ISA requirement).
    hs = __builtin_amdgcn_readfirstlane(hs);
    he = __builtin_amdgcn_readfirstlane(he);
    ws = __builtin_amdgcn_readfirstlane(ws);
    we = __builtin_amdgcn_readfirstlane(we);

    // A-matrix lane mapping for 16x16x4 f32:
    //   lanes 0-15  hold (K=0,K=1) of row M=lane
    //   lanes 16-31 hold (K=2,K=3) of row M=lane-16
    const int m    = lane & 15;
    const int koff = (lane >> 4) << 1;           // 0 or 2
    const int ch0  = m;
    const int ch1  = min(16 + m, C_OUT - 1);     // clamp: rows 5..15 are dead

    // 32-bit element offsets from the uniform base pointer (tensor is
    // 37.9M elements, fits easily in i32) -> SADDR + 32-bit VADDR loads.
    const int plane = i * PWW + j;
    const int base0 = (b * CIN + ch0 * (PHH * PWW) + plane) * (HH * WW);
    const int base1 = (b * CIN + ch1 * (PHH * PWW) + plane) * (HH * WW);

    v8f acc0 = {};   // channels 0..15
    v8f acc1 = {};   // channels 16..20 in rows 0..4; rows 5..15 unused
    v2f ones; ones.x = 1.0f; ones.y = 1.0f;

    for (int y = hs; y < he; ++y) {
        const int row = y * WW;
        for (int xb = ws; xb < we; xb += 4) {
            const int c0  = xb + koff;
            const int c1  = c0 + 1;
            const int c0c = min(c0, we - 1);     // clamped, always in range
            const int c1c = min(c1, we - 1);
            // Four unconditional loads (clamped addresses are always valid).
            const float v00 = x[base0 + row + c0c];
            const float v01 = x[base0 + row + c1c];
            const float v10 = x[base1 + row + c0c];
            const float v11 = x[base1 + row + c1c];
            // Zero out-of-range K slots via selects (2 compares, 4 cndmasks).
            v2f a0, a1;
            a0.x = (c0 < we) ? v00 : 0.0f;
            a0.y = (c1 < we) ? v01 : 0.0f;
            a1.x = (c0 < we) ? v10 : 0.0f;
            a1.y = (c1 < we) ? v11 : 0.0f;
            // D = A x ones + C : every D column = per-channel partial sum.
            acc0 = __builtin_amdgcn_wmma_f32_16x16x4_f32(
                       false, a0, false, ones, (short)0, acc0, false, false);
            acc1 = __builtin_amdgcn_wmma_f32_16x16x4_f32(
                       false, a1, false, ones, (short)0, acc1, false, false);
        }
    }

    // Column 0 of D lives in lane 0 (rows 0-7) and lane 16 (rows 8-15),
    // one row per C/D VGPR. Bounce through LDS to fan out to lanes 0..20.
    __shared__ float sums[32];
    if (lane == 0) {
#pragma unroll
        for (int r = 0; r < 8; ++r) sums[r] = acc0[r];
#pragma unroll
        for (int r = 0; r < 5; ++r) sums[16 + r] = acc1[r];
    }
    if (lane == 16) {
#pragma unroll
        for (int r = 0; r < 8; ++r) sums[8 + r] = acc0[r];
    }
    __syncthreads();

    const int area = (he - hs) * (we - ws);
    if (lane < C_OUT) {
        const float v = (area > 0) ? sums[lane] / (float)area : 0.0f;
        out[((size_t)k * C_OUT + lane) * (PHH * PWW) + bin] = v;
    }
}

extern "C" void kernel_launch(void* const* d_in, const int* in_sizes, int n_in,
                              void* d_out, int out_size, void* d_ws, size_t ws_size,
                              hipStream_t stream) {
    const float* x    = (const float*)d_in[0];   // (4, 1029, 96, 96) f32
    const float* rois = (const float*)d_in[1];   // (512, 5) f32
    float*       out  = (float*)d_out;           // (512, 21, 7, 7) f32

    const int K = in_sizes[1] / 5;
    dim3 grid(PHH * PWW, K);
    psroi_pool_wmma<<<grid, 32, 0, stream>>>(x, rois, out);
}